// AnchornizedNMS_85194971283814
// MI455X (gfx1250) — compile-verified
//
#include <hip/hip_runtime.h>
#include <hip/hip_bf16.h>
#include <stdint.h>

// ---------------- problem constants (match reference) ----------------
#define NCLS    80
#define CDIM    85          // x,y,w,h,obj + 80 class scores
#define NBOX    25200
#define NB      16
#define KCAP    4096        // NMS candidate count (reference K)
#define CANDCAP 8192        // oversized candidate pool (power of two for bitonic)
#define MAXDET  300
#define CONF_TH 0.6f
#define IOU_TH  0.45f
#define MAX_WH  7680.0f
#define HBINS   1024

// ---------------- CDNA5 async / TDM helpers ----------------
#if defined(__has_builtin)
# if __has_builtin(__builtin_amdgcn_global_load_async_to_lds_b32)
#  define HAVE_ASYNC_LDS 1
# endif
# if __has_builtin(__builtin_amdgcn_s_wait_asynccnt)
#  define HAVE_WAIT_ASYNC 1
# endif
# if __has_builtin(__builtin_amdgcn_tensor_load_to_lds)
#  define HAVE_TDM 1
# endif
# if __has_builtin(__builtin_amdgcn_s_wait_tensorcnt)
#  define HAVE_WAIT_TENSOR 1
# endif
#endif

typedef __attribute__((address_space(1))) int g1_int;   // global
typedef __attribute__((address_space(3))) int l3_int;   // LDS

typedef unsigned int v4u __attribute__((ext_vector_type(4)));
typedef int v8i __attribute__((ext_vector_type(8)));
typedef int v4i __attribute__((ext_vector_type(4)));

__device__ __forceinline__ void async_copy_dword_to_lds(const float* g, float* l) {
#if defined(HAVE_ASYNC_LDS)
  // emits global_load_async_to_lds_b32 (tracked by ASYNCcnt)
  __builtin_amdgcn_global_load_async_to_lds_b32(
      (g1_int*)(void*)const_cast<float*>(g), (l3_int*)(void*)l, 0, 0);
#else
  asm volatile("global_load_async_to_lds_b32 %0, %1, off"
               :: "v"((unsigned)(uintptr_t)l), "v"(g) : "memory");
#endif
}

__device__ __forceinline__ void wait_async_zero() {
#if defined(HAVE_WAIT_ASYNC)
  __builtin_amdgcn_s_wait_asynccnt(0);
#else
  asm volatile("s_wait_asynccnt 0" ::: "memory");
#endif
}

__device__ __forceinline__ void wait_tensor_zero() {
#if defined(HAVE_WAIT_TENSOR)
  __builtin_amdgcn_s_wait_tensorcnt(0);
#else
  asm volatile("s_wait_tensorcnt 0" ::: "memory");
#endif
}

// LDS byte offset of a shared-memory object (AS3 pointers are 32-bit offsets)
__device__ __forceinline__ unsigned lds_offset_of(void* p) {
  return (unsigned)(uintptr_t)(l3_int*)p;
}

// ---------------- kernel 1: zero scratch counters ----------------
__global__ void k_init(uint32_t* __restrict__ hist, uint32_t* __restrict__ cnt,
                       uint32_t* __restrict__ thrbin) {
  int t = blockIdx.x * blockDim.x + threadIdx.x;
  if (t < NB * HBINS) hist[t] = 0u;
  if (t < NB) { cnt[t] = 0u; thrbin[t] = 0u; }
}

// ---------------- kernel 2: per-image objness histogram ----------------
// obj values staged through LDS via the async-copy engine (ASYNCcnt domain)
__global__ __launch_bounds__(256)
void k_hist(const float* __restrict__ x, uint32_t* __restrict__ hist) {
  __shared__ float sobj[256];
  int b = blockIdx.y;
  int i = blockIdx.x * 256 + (int)threadIdx.x;
  if (i < NBOX)
    async_copy_dword_to_lds(&x[((size_t)b * NBOX + i) * CDIM + 4],
                            &sobj[threadIdx.x]);
  wait_async_zero();
  __syncthreads();
  if (i >= NBOX) return;
  float obj = sobj[threadIdx.x];
  if (obj > CONF_TH) {
    int bin = (int)((obj - CONF_TH) * ((float)HBINS / (1.0f - CONF_TH)));
    bin = bin < 0 ? 0 : (bin > HBINS - 1 ? HBINS - 1 : bin);
    atomicAdd(&hist[b * HBINS + bin], 1u);
  }
}

// ---------------- kernel 3: obj cutoff bin for top-K-by-obj ----------------
__global__ void k_thresh(const uint32_t* __restrict__ hist,
                         uint32_t* __restrict__ thrbin) {
  int b = blockIdx.x;
  if (threadIdx.x == 0) {
    uint32_t total = 0; int t = 0;
    for (int bin = HBINS - 1; bin >= 0; --bin) {
      total += hist[b * HBINS + bin];
      if (total >= KCAP) { t = bin; break; }
    }
    thrbin[b] = (uint32_t)t;
  }
}

// ---------------- kernel 4: TDM-staged candidate build ----------------
#define TILE 128
__global__ __launch_bounds__(TILE)
void k_build(const float* __restrict__ x, const uint32_t* __restrict__ thrbin,
             uint32_t* __restrict__ cnt, float* __restrict__ cand /* NB*CANDCAP*6 */) {
  __shared__ float lds[TILE * CDIM];                 // 43.5 KB tile
  int b    = blockIdx.y;
  int row0 = blockIdx.x * TILE;
  int rows = NBOX - row0; if (rows > TILE) rows = TILE;
  const float* src = x + ((size_t)b * NBOX + row0) * CDIM;

#if defined(HAVE_TDM)
  // One Tensor-Data-Mover DMA per block: 2D tile, rows x 85 f32, stride 85.
  if (threadIdx.x < 32) {     // wave 0 issues the descriptor (TDM ignores EXEC)
    uint64_t ga = (uint64_t)(uintptr_t)src;
    unsigned la = lds_offset_of((void*)lds);
    v4u g0;
    g0.x = 1u;                                          // count=1 (valid user D#)
    g0.y = la;                                          // lds_addr
    g0.z = (unsigned)(ga & 0xFFFFFFFFu);                // global_addr[31:0]
    g0.w = (unsigned)((ga >> 32) & 0x01FFFFFFu)         // global_addr[56:32]
         | 0x80000000u;                                 // type=2 ("image")
    v8i g1;
    g1[0] = (int)(2u << 16);                            // data_size=4B, no flags
    g1[1] = (int)(((unsigned)CDIM & 0xFFFFu) << 16);    // tensor_dim0[15:0]=85
    g1[2] = (int)(((unsigned)rows & 0xFFFFu) << 16);    // dim0 hi=0 | tensor_dim1 lo=rows
    g1[3] = (int)(((unsigned)CDIM & 0xFFFFu) << 16);    // dim1 hi=0 | tile_dim0=85
    g1[4] = (int)((unsigned)rows & 0xFFFFu);            // tile_dim1=rows, tile_dim2=0
    g1[5] = (int)CDIM;                                  // tensor_dim0_stride lo32 = 85
    g1[6] = 0;                                          // stride hi | dim1_stride lo
    g1[7] = 0;                                          // dim1_stride hi
    v4i z4 = {0, 0, 0, 0};                              // 2D tensor: groups 2/3 unused
    v8i z8 = {0, 0, 0, 0, 0, 0, 0, 0};
    __builtin_amdgcn_tensor_load_to_lds(g0, g1, z4, z4, z8, 0);
  }
  wait_tensor_zero();   // no-op for non-issuing waves (their TENSORcnt==0)
  __syncthreads();
#else
  int total = rows * CDIM;
  for (int idx = threadIdx.x; idx < total; idx += TILE)
    async_copy_dword_to_lds(src + idx, &lds[idx]);
  wait_async_zero();
  __syncthreads();
#endif

  int r = threadIdx.x;
  if (r < rows) {
    const float* p = &lds[r * CDIM];
    float obj = p[4];
    if (obj > CONF_TH) {
      int bin = (int)((obj - CONF_TH) * ((float)HBINS / (1.0f - CONF_TH)));
      bin = bin < 0 ? 0 : (bin > HBINS - 1 ? HBINS - 1 : bin);
      if ((uint32_t)bin >= thrbin[b]) {
        float best = p[5]; int bc = 0;
        #pragma unroll 4
        for (int c = 1; c < NCLS; ++c) {
          float v = p[5 + c];
          if (v > best) { best = v; bc = c; }
        }
        uint32_t pos = atomicAdd(&cnt[b], 1u);
        if (pos < CANDCAP) {
          float cx = p[0], cy = p[1], hw = p[2] * 0.5f, hh = p[3] * 0.5f;
          float* o = cand + ((size_t)b * CANDCAP + pos) * 6;
          o[0] = cx - hw; o[1] = cy - hh; o[2] = cx + hw; o[3] = cy + hh;
          o[4] = obj * best; o[5] = (float)bc;
        }
      }
    }
  }
}

// ---------------- kernel 5: per-image bitonic sort by score ----------------
__global__ __launch_bounds__(1024)
void k_sort(const float* __restrict__ cand, const uint32_t* __restrict__ cnt,
            float* __restrict__ sbox, float* __restrict__ sscore,
            float* __restrict__ scls) {
  __shared__ uint32_t skey[CANDCAP];   // 32 KB
  __shared__ uint16_t sval[CANDCAP];   // 16 KB
  int b = blockIdx.x;
  uint32_t n = cnt[b]; if (n > CANDCAP) n = CANDCAP;
  const uint32_t INVK = 0xFFFFFFFFu;

  for (int r = threadIdx.x; r < CANDCAP; r += 1024) {
    uint32_t key = INVK; uint16_t v = 0;
    if ((uint32_t)r < n) {
      float s = cand[((size_t)b * CANDCAP + r) * 6 + 4];   // score >= 0
      key = ~(__float_as_uint(s) | 0x80000000u);           // ascending => score desc
      v = (uint16_t)r;
    }
    skey[r] = key; sval[r] = v;
  }
  __syncthreads();

  for (int k = 2; k <= CANDCAP; k <<= 1) {
    for (int j = k >> 1; j > 0; j >>= 1) {
      for (int t = threadIdx.x; t < CANDCAP; t += 1024) {
        int ixj = t ^ j;
        if (ixj > t) {
          bool up = ((t & k) == 0);
          uint32_t a = skey[t], c = skey[ixj];
          if ((a > c) == up) {
            skey[t] = c; skey[ixj] = a;
            uint16_t tv = sval[t]; sval[t] = sval[ixj]; sval[ixj] = tv;
          }
        }
      }
      __syncthreads();
    }
  }

  // emit top KCAP in score-descending order, SoA
  for (int r = threadIdx.x; r < KCAP; r += 1024) {
    size_t o = (size_t)b * KCAP + r;
    uint32_t key = skey[r];
    if (key != INVK) {
      const float* c = cand + ((size_t)b * CANDCAP + sval[r]) * 6;
      sbox[o * 4 + 0] = c[0]; sbox[o * 4 + 1] = c[1];
      sbox[o * 4 + 2] = c[2]; sbox[o * 4 + 3] = c[3];
      sscore[o] = c[4]; scls[o] = c[5];
    } else {
      sbox[o * 4 + 0] = 0.f; sbox[o * 4 + 1] = 0.f;
      sbox[o * 4 + 2] = 0.f; sbox[o * 4 + 3] = 0.f;
      sscore[o] = -1.0f; scls[o] = 0.f;
    }
  }
}

// ---------------- kernel 6: greedy NMS + top-300 compaction ----------------
__global__ __launch_bounds__(1024)
void k_nms_out(const float* __restrict__ sbox, const float* __restrict__ sscore,
               const float* __restrict__ scls, float* __restrict__ out) {
  __shared__ unsigned char keep[KCAP];   // 4 KB
  __shared__ float bc[5];
  __shared__ unsigned int ssum[1024];
  int b = blockIdx.x, tid = threadIdx.x;

  // 4 class-offset boxes per thread, strided ownership r = tid + s*1024
  float bx1[4], by1[4], bx2[4], by2[4], area[4];
  bool kp[4];
  #pragma unroll
  for (int s = 0; s < 4; ++s) {
    int r = tid + s * 1024;
    size_t o = (size_t)b * KCAP + r;
    float off = scls[o] * MAX_WH;
    float x1 = sbox[o * 4 + 0] + off, y1 = sbox[o * 4 + 1] + off;
    float x2 = sbox[o * 4 + 2] + off, y2 = sbox[o * 4 + 3] + off;
    bx1[s] = x1; by1[s] = y1; bx2[s] = x2; by2[s] = y2;
    area[s] = (x2 - x1) * (y2 - y1);
    kp[s] = (sscore[o] >= 0.0f);
    keep[r] = kp[s] ? 1 : 0;
  }
  __syncthreads();

  // sequential pivot scan (exact greedy NMS order)
  for (int i = 0; i < KCAP; ++i) {
    if (!keep[i]) continue;                 // uniform across workgroup
    int oi = i & 1023, si = i >> 10;
    if (tid == oi) {
      bc[0] = bx1[si]; bc[1] = by1[si]; bc[2] = bx2[si]; bc[3] = by2[si];
      bc[4] = area[si];
    }
    __syncthreads();
    float cx1 = bc[0], cy1 = bc[1], cx2 = bc[2], cy2 = bc[3], ca = bc[4];
    #pragma unroll
    for (int s = 0; s < 4; ++s) {
      int r = tid + s * 1024;
      if (r > i && kp[s]) {
        float iw = fminf(bx2[s], cx2) - fmaxf(bx1[s], cx1);
        float ih = fminf(by2[s], cy2) - fmaxf(by1[s], cy1);
        iw = fmaxf(iw, 0.0f); ih = fmaxf(ih, 0.0f);
        float inter = iw * ih;
        float iou = inter / (area[s] + ca - inter + 1e-9f);
        if (iou > IOU_TH) { kp[s] = false; keep[r] = 0; }
      }
    }
    __syncthreads();
  }

  // prefix-scan keep flags (contiguous 4 per thread) -> output ranks
  int base = tid * 4;
  int loc[4]; unsigned tsum = 0;
  #pragma unroll
  for (int k = 0; k < 4; ++k) { loc[k] = keep[base + k]; tsum += (unsigned)loc[k]; }
  ssum[tid] = tsum;
  __syncthreads();
  for (int ofs = 1; ofs < 1024; ofs <<= 1) {
    unsigned add = (tid >= ofs) ? ssum[tid - ofs] : 0u;
    __syncthreads();
    ssum[tid] += add;
    __syncthreads();
  }
  unsigned incl  = ssum[tid];
  unsigned total = ssum[1023];
  unsigned rank  = incl - tsum;

  float* valid_out = out + (size_t)NB * MAXDET * 6;
  #pragma unroll
  for (int k = 0; k < 4; ++k) {
    if (loc[k]) {
      if (rank < MAXDET) {
        int r = base + k;
        size_t o = (size_t)b * KCAP + r;
        float sc = sscore[o];
        float v = (sc > 0.0f) ? 1.0f : 0.0f;
        float* orow = out + ((size_t)b * MAXDET + rank) * 6;
        orow[0] = sbox[o * 4 + 0] * v; orow[1] = sbox[o * 4 + 1] * v;
        orow[2] = sbox[o * 4 + 2] * v; orow[3] = sbox[o * 4 + 3] * v;
        orow[4] = sc * v;              orow[5] = scls[o] * v;
        valid_out[(size_t)b * MAXDET + rank] = v;
      }
      rank++;
    }
  }
  unsigned tc = total < MAXDET ? total : (unsigned)MAXDET;
  for (int o2 = tid; o2 < MAXDET; o2 += 1024) {
    if ((unsigned)o2 >= tc) {
      float* orow = out + ((size_t)b * MAXDET + o2) * 6;
      orow[0] = orow[1] = orow[2] = orow[3] = orow[4] = orow[5] = 0.0f;
      valid_out[(size_t)b * MAXDET + o2] = 0.0f;
    }
  }
}

// ---------------- launcher ----------------
extern "C" void kernel_launch(void* const* d_in, const int* in_sizes, int n_in,
                              void* d_out, int out_size, void* d_ws, size_t ws_size,
                              hipStream_t stream) {
  const float* x = (const float*)d_in[0];
  float* out = (float*)d_out;

  uint8_t* ws = (uint8_t*)d_ws;
  size_t off = 0;
  auto carve = [&](size_t bytes) -> void* {
    void* p = ws + off;
    off += (bytes + 255) & ~(size_t)255;
    return p;
  };
  uint32_t* hist   = (uint32_t*)carve((size_t)NB * HBINS * sizeof(uint32_t));
  uint32_t* cnt    = (uint32_t*)carve((size_t)NB * sizeof(uint32_t));
  uint32_t* thrbin = (uint32_t*)carve((size_t)NB * sizeof(uint32_t));
  float*    cand   = (float*)carve((size_t)NB * CANDCAP * 6 * sizeof(float));
  float*    sbox   = (float*)carve((size_t)NB * KCAP * 4 * sizeof(float));
  float*    sscore = (float*)carve((size_t)NB * KCAP * sizeof(float));
  float*    sclsb  = (float*)carve((size_t)NB * KCAP * sizeof(float));
  (void)ws_size; (void)in_sizes; (void)n_in; (void)out_size;

  k_init  <<<dim3((NB * HBINS + 255) / 256), 256, 0, stream>>>(hist, cnt, thrbin);
  k_hist  <<<dim3((NBOX + 255) / 256, NB), 256, 0, stream>>>(x, hist);
  k_thresh<<<dim3(NB), 32, 0, stream>>>(hist, thrbin);
  k_build <<<dim3((NBOX + TILE - 1) / TILE, NB), TILE, 0, stream>>>(x, thrbin, cnt, cand);
  k_sort  <<<dim3(NB), 1024, 0, stream>>>(cand, cnt, sbox, sscore, sclsb);
  k_nms_out<<<dim3(NB), 1024, 0, stream>>>(sbox, sscore, sclsb, out);
}